// LrDistance_15015205667395
// MI455X (gfx1250) — compile-verified
//
#include <hip/hip_runtime.h>
#include <stdint.h>

// Problem shape (fixed by the reference's setup_inputs).
constexpr int S = 16;
constexpr int M = 768;
constexpr int N = 1024;
constexpr float INVALID_VAL = 100.0f;

// align_corners=False sample-coordinate scales, folded to a single FMA:
//   ix = ((2*xr/(N-1) + 1)*N - 1)/2  ==  xr * (N/(N-1)) - 0.5
//   iy = ((2*y /(M-1) + 1)*M - 1)/2  ==  y  * (M/(M-1)) - 0.5
constexpr float SX = (float)((double)N / (double)(N - 1));
constexpr float SY = (float)((double)M / (double)(M - 1));

// One block per (image s, output row y). 256 threads (8 wave32), 4 px/thread.
// The two disps_rl source rows needed by this output row are staged into LDS
// with CDNA5 async global->LDS copies (ASYNCcnt-tracked), overlapped with the
// regular load of the disps_lr row.
__global__ __launch_bounds__(256) void lr_distance_kernel(
    const float* __restrict__ disps_lr,
    const float* __restrict__ disps_rl,
    float* __restrict__ out)
{
    __shared__ float row0[N];
    __shared__ float row1[N];

    const int y = blockIdx.x;   // 0..M-1
    const int s = blockIdx.y;   // 0..S-1
    const int t = threadIdx.x;  // 0..255
    const int col = t * 4;      // 4 floats (16B) per lane -> 1024 floats/row

    const float* lr_row = disps_lr + ((size_t)s * M + y) * (size_t)N;
    const float* rl_img = disps_rl + (size_t)s * M * (size_t)N;
    float*       o_row  = out      + ((size_t)s * M + y) * (size_t)N;

    // Vertical sample position (per-row constant): one FMA instead of an IEEE
    // divide-by-constant sequence.
    const float iy  = fmaf((float)y, SY, -0.5f);
    const float y0f = floorf(iy);
    const float wy1 = iy - y0f;
    const float wy0 = 1.0f - wy1;
    const int   y0  = (int)y0f;
    const int   y1  = y0 + 1;
    const bool  vy0 = (y0 >= 0) && (y0 <= M - 1);
    const bool  vy1 = (y1 >= 0) && (y1 <= M - 1);

    // --- CDNA5 async global->LDS staging of the two source rows -------------
    // vdst VGPR carries the wave-relative LDS byte address (low 32 bits of the
    // generic shared pointer); vaddr is the 64-bit global address.
    {
        unsigned l0 = (unsigned)(uintptr_t)(&row0[col]);
        unsigned l1 = (unsigned)(uintptr_t)(&row1[col]);
        if (vy0) {
            const float* g = rl_img + (size_t)y0 * N + col;
            asm volatile("global_load_async_to_lds_b128 %0, %1, off"
                         :: "v"(l0), "v"(g) : "memory");
        }
        if (vy1) {
            const float* g = rl_img + (size_t)y1 * N + col;
            asm volatile("global_load_async_to_lds_b128 %0, %1, off"
                         :: "v"(l1), "v"(g) : "memory");
        }
    }

    // Overlap: pull this row's disparities while the async copies stream.
    const float4 d4 = *(const float4*)(lr_row + col);
    const float dv[4] = {d4.x, d4.y, d4.z, d4.w};

    asm volatile("s_wait_asynccnt 0x0" ::: "memory");
    __syncthreads();

    // --- bilinear warp + LR consistency ------------------------------------
    float r[4];
#pragma unroll
    for (int k = 0; k < 4; ++k) {
        const int   x  = col + k;
        const float d  = dv[k];
        const float xr = (float)x - d;
        const bool  invalid = (xr >= (float)N) || (xr < 0.0f);

        // Horizontal sample position: single FMA (see SX derivation above).
        const float ix  = fmaf(xr, SX, -0.5f);
        const float x0f = floorf(ix);
        const float wx1 = ix - x0f;
        const float wx0 = 1.0f - wx1;

        const int  x0  = (int)x0f;
        const int  x1  = x0 + 1;
        const bool vx0 = (x0 >= 0) && (x0 <= N - 1);
        const bool vx1 = (x1 >= 0) && (x1 <= N - 1);
        const int  c0  = min(max(x0, 0), N - 1);
        const int  c1  = min(max(x1, 0), N - 1);

        const float v00 = (vy0 && vx0) ? row0[c0] : 0.0f;
        const float v01 = (vy0 && vx1) ? row0[c1] : 0.0f;
        const float v10 = (vy1 && vx0) ? row1[c0] : 0.0f;
        const float v11 = (vy1 && vx1) ? row1[c1] : 0.0f;

        // Factor the wave-uniform vertical weights out of the inner products.
        const float h0 = fmaf(v00, wx0, v01 * wx1);
        const float h1 = fmaf(v10, wx0, v11 * wx1);
        const float warped = fmaf(h0, wy0, h1 * wy1);

        r[k] = invalid ? INVALID_VAL : fabsf(d + warped);
    }

    *(float4*)(o_row + col) = make_float4(r[0], r[1], r[2], r[3]);
}

extern "C" void kernel_launch(void* const* d_in, const int* in_sizes, int n_in,
                              void* d_out, int out_size, void* d_ws, size_t ws_size,
                              hipStream_t stream) {
    (void)in_sizes; (void)n_in; (void)out_size; (void)d_ws; (void)ws_size;
    const float* disps_lr = (const float*)d_in[0];
    const float* disps_rl = (const float*)d_in[1];
    float*       out      = (float*)d_out;

    dim3 grid(M, S, 1);   // one block per (row, image)
    dim3 block(256, 1, 1);
    lr_distance_kernel<<<grid, block, 0, stream>>>(disps_lr, disps_rl, out);
}